// genie_mmse_estimator_77790447665766
// MI455X (gfx1250) — compile-verified
//
#include <hip/hip_runtime.h>
#include <stdint.h>

typedef float v2f  __attribute__((ext_vector_type(2)));
typedef float v8f  __attribute__((ext_vector_type(8)));
typedef unsigned int u32x4 __attribute__((ext_vector_type(4)));
typedef int   i32x4 __attribute__((ext_vector_type(4)));
typedef int   i32x8 __attribute__((ext_vector_type(8)));

static constexpr int NN  = 64;   // matrix dim
static constexpr int KK  = 14;   // RHS count
static constexpr int KP  = 16;   // padded RHS count
static constexpr int LDA = 65;   // LDS row stride (floats) -> bank-conflict-free columns

// Low 32 bits of a generic pointer to LDS == raw LDS byte offset.
__device__ __forceinline__ unsigned lds_off(const void* p) {
  return (unsigned)(uintptr_t)p;
}

// TDM: async DMA of a 64x64 f32 row-major tile (global stride = 64) into LDS,
// inserting 1 DWORD of padding after every 64 DWORDs -> LDS row stride 65 floats.
__device__ __forceinline__ void tdm_load_2d(unsigned lds_byte_addr, const float* gptr) {
  unsigned long long ga = (unsigned long long)(uintptr_t)gptr;
  u32x4 g0;
  g0.x = 1u;                                    // count=1, user descriptor
  g0.y = lds_byte_addr;                         // LDS destination
  g0.z = (unsigned)(ga & 0xFFFFFFFFu);          // global_addr[31:0]
  g0.w = (unsigned)((ga >> 32) & 0x1FFFFFFu) | (2u << 30);   // global_addr[56:32] | type=2
  i32x8 g1;
  // data_size=4B (2), pad_enable, pad_interval=5 (64 DWORDs), pad_amount=0 (1 DWORD)
  g1[0] = (2 << 16) | (1 << 20) | (5 << 22);
  g1[1] = (NN & 0xFFFF) << 16;                  // tensor_dim0[15:0]
  g1[2] = ((NN >> 16) & 0xFFFF) | ((NN & 0xFFFF) << 16);   // dim0 hi | tensor_dim1 lo
  g1[3] = ((NN >> 16) & 0xFFFF) | ((NN & 0xFFFF) << 16);   // dim1 hi | tile_dim0
  g1[4] = (NN & 0xFFFF);                        // tile_dim1 (tile_dim2 = 0)
  g1[5] = NN;                                   // tensor_dim0_stride[31:0]
  g1[6] = 0;                                    // stride0 hi | stride1 lo
  g1[7] = 0;                                    // stride1 hi
  i32x4 g2 = {0, 0, 0, 0};
  i32x4 g3 = {0, 0, 0, 0};
  i32x8 g4 = {0, 0, 0, 0, 0, 0, 0, 0};          // 6-arg toolchain variant: extra group
  __builtin_amdgcn_tensor_load_to_lds(g0, g1, g2, g3, g4, 0);
}

__global__ __launch_bounds__(256)
void mmse_kernel(const float* __restrict__ y_r, const float* __restrict__ y_i,
                 const float* __restrict__ C_r, const float* __restrict__ C_i,
                 const float* __restrict__ no_p, const float* __restrict__ pr_p,
                 const float* __restrict__ pi_p, float* __restrict__ out) {
  __shared__ float Ar[NN * LDA];
  __shared__ float Ai[NN * LDA];
  __shared__ float Zr[NN * KP];
  __shared__ float Zi[NN * KP];

  const int t    = threadIdx.x;
  const int b    = blockIdx.x;
  const int wave = t >> 5;
  const int lane = t & 31;
  const int lm   = lane & 15;
  const int lh   = lane >> 4;

  const float no = no_p[0];
  const float pr = pr_p[0];
  const float pi = pi_p[0];
  const float s  = pr * pr + pi * pi;           // |pilot|^2

  const float* Crb = C_r + (size_t)b * NN * NN;
  const float* Cib = C_i + (size_t)b * NN * NN;
  const float* yrb = y_r + (size_t)b * KK * NN;
  const float* yib = y_i + (size_t)b * KK * NN;

  // ---- TDM async loads of C planes into Ar/Ai (wave 0 only; TDM ignores EXEC) ----
  if (wave == 0) {
    tdm_load_2d(lds_off(Ar), Crb);
    tdm_load_2d(lds_off(Ai), Cib);
    __builtin_amdgcn_s_wait_tensorcnt(0);
  }
  // Meanwhile all threads stage Z = y^T (pad cols 14..15 with zeros)
  for (int idx = t; idx < NN * KP; idx += 256) {
    int n = idx / KP, k = idx % KP;
    float vr = 0.f, vi = 0.f;
    if (k < KK) { vr = yrb[k * NN + n]; vi = yib[k * NN + n]; }
    Zr[idx] = vr; Zi[idx] = vi;
  }
  __syncthreads();

  // ---- A = s*C + no*I (Hermitian positive definite) ----
  for (int idx = t; idx < NN * NN; idx += 256) {
    int i = idx >> 6, j = idx & 63;
    float ar = Ar[i * LDA + j] * s;
    float ai = Ai[i * LDA + j] * s;
    if (i == j) { ar += no; ai = 0.f; }
    Ar[i * LDA + j] = ar;
    Ai[i * LDA + j] = ai;
  }
  __syncthreads();

  // ---- Blocked (16) right-looking complex Cholesky: A = L*L^H, L in lower ----
  for (int jb = 0; jb < 4; ++jb) {
    const int j0 = jb * 16, jend = j0 + 16;

    // Panel factor: unblocked over 16 columns, updates restricted to the panel
    for (int j = j0; j < jend; ++j) {
      float d    = Ar[j * LDA + j];
      float dinv = rsqrtf(d);
      for (int i = j + t; i < NN; i += 256) {
        if (i == j) { Ar[j * LDA + j] = d * dinv; Ai[j * LDA + j] = 0.f; }
        else        { Ar[i * LDA + j] *= dinv;    Ai[i * LDA + j] *= dinv; }
      }
      __syncthreads();
      int ncols = jend - 1 - j;
      for (int idx = t; idx < ncols * NN; idx += 256) {
        int c = j + 1 + idx / NN;
        int i = idx % NN;
        if (i >= c) {
          float lir = Ar[i * LDA + j], lii = Ai[i * LDA + j];
          float lcr = Ar[c * LDA + j], lci = Ai[c * LDA + j];
          Ar[i * LDA + c] -= lir * lcr + lii * lci;   // -= Re(L[i]*conj(L[c]))
          Ai[i * LDA + c] -= lii * lcr - lir * lci;   // -= Im(L[i]*conj(L[c]))
        }
      }
      __syncthreads();
    }

    // Trailing update S -= P*P^H via f32 WMMA (lower-triangle 16x16 tiles, 1 per wave)
    int rem = NN - jend;
    if (rem > 0) {
      int nt = rem >> 4;
      int ntiles = nt * (nt + 1) / 2;
      for (int tid = wave; tid < ntiles; tid += 8) {
        int ti = 0, tj = 0;
        { int s2 = tid, r = 0;
          while (s2 > r) { s2 -= r + 1; ++r; }
          ti = r; tj = s2; }
        const int r0 = jend + ti * 16;   // tile rows
        const int c0 = jend + tj * 16;   // tile cols

        // Seed accumulators with the live S tile (C/D layout: m = r + 8*(lane/16), n = lane%16)
        v8f accr, acci;
        for (int r = 0; r < 8; ++r) {
          int m = r + 8 * lh;
          accr[r] = Ar[(r0 + m) * LDA + (c0 + lm)];
          acci[r] = Ai[(r0 + m) * LDA + (c0 + lm)];
        }
        for (int kc = 0; kc < 4; ++kc) {
          // A/B operand k index: k = 2*(lane/16) + v  (v = vector element)
          int kg = j0 + kc * 4 + 2 * lh;
          v2f arp, aip, brp, bip;
          arp[0] = Ar[(r0 + lm) * LDA + kg];  arp[1] = Ar[(r0 + lm) * LDA + kg + 1];
          aip[0] = Ai[(r0 + lm) * LDA + kg];  aip[1] = Ai[(r0 + lm) * LDA + kg + 1];
          brp[0] = Ar[(c0 + lm) * LDA + kg];  brp[1] = Ar[(c0 + lm) * LDA + kg + 1];
          bip[0] = Ai[(c0 + lm) * LDA + kg];  bip[1] = Ai[(c0 + lm) * LDA + kg + 1];
          v2f narp = -arp, naip = -aip;       // f32 WMMA has no A/B NEG -> negate in regs
          // S_r += (-Pr)Pr^T + (-Pi)Pi^T
          accr = __builtin_amdgcn_wmma_f32_16x16x4_f32(false, narp, false, brp, (short)0, accr, false, false);
          accr = __builtin_amdgcn_wmma_f32_16x16x4_f32(false, naip, false, bip, (short)0, accr, false, false);
          // S_i += (-Pi)Pr^T + ( Pr)Pi^T
          acci = __builtin_amdgcn_wmma_f32_16x16x4_f32(false, naip, false, brp, (short)0, acci, false, false);
          acci = __builtin_amdgcn_wmma_f32_16x16x4_f32(false, arp,  false, bip, (short)0, acci, false, false);
        }
        for (int r = 0; r < 8; ++r) {
          int m = r + 8 * lh;
          Ar[(r0 + m) * LDA + (c0 + lm)] = accr[r];
          Ai[(r0 + m) * LDA + (c0 + lm)] = acci[r];
        }
      }
    }
    __syncthreads();
  }

  // ---- Forward solve: L * W = Z (in place, 16 padded RHS) ----
  for (int j = 0; j < NN; ++j) {
    float dinv = 1.0f / Ar[j * LDA + j];
    if (t < KP) { Zr[j * KP + t] *= dinv; Zi[j * KP + t] *= dinv; }
    __syncthreads();
    for (int idx = t; idx < (NN - 1 - j) * KP; idx += 256) {
      int i = j + 1 + idx / KP;
      int k = idx % KP;
      float lr = Ar[i * LDA + j], li = Ai[i * LDA + j];
      float zr = Zr[j * KP + k],  zi = Zi[j * KP + k];
      Zr[i * KP + k] -= lr * zr - li * zi;
      Zi[i * KP + k] -= lr * zi + li * zr;
    }
    __syncthreads();
  }

  // ---- Backward solve: L^H * Z = W (in place) ----
  for (int j = NN - 1; j >= 0; --j) {
    float dinv = 1.0f / Ar[j * LDA + j];
    if (t < KP) { Zr[j * KP + t] *= dinv; Zi[j * KP + t] *= dinv; }
    __syncthreads();
    for (int idx = t; idx < j * KP; idx += 256) {
      int i = idx / KP;
      int k = idx % KP;
      float lr = Ar[j * LDA + i], li = -Ai[j * LDA + i];   // conj(L[j][i])
      float zr = Zr[j * KP + k],  zi = Zi[j * KP + k];
      Zr[i * KP + k] -= lr * zr - li * zi;
      Zi[i * KP + k] -= lr * zi + li * zr;
    }
    __syncthreads();
  }

  // ---- Epilogue: out[b,k,n] = (conj(p)/s) * (y[b,k,n] - no * z[n,k]), complex64 interleaved ----
  const float cpr = pr / s, cpi = -pi / s;
  float* ob = out + (size_t)b * KK * NN * 2;
  for (int idx = t; idx < KK * NN; idx += 256) {
    int k = idx / NN, n = idx % NN;
    float vr = yrb[idx] - no * Zr[n * KP + k];
    float vi = yib[idx] - no * Zi[n * KP + k];
    ob[idx * 2 + 0] = cpr * vr - cpi * vi;
    ob[idx * 2 + 1] = cpr * vi + cpi * vr;
  }
}

extern "C" void kernel_launch(void* const* d_in, const int* in_sizes, int n_in,
                              void* d_out, int out_size, void* d_ws, size_t ws_size,
                              hipStream_t stream) {
  (void)n_in; (void)out_size; (void)d_ws; (void)ws_size;
  const float* y_r = (const float*)d_in[0];
  const float* y_i = (const float*)d_in[1];
  const float* C_r = (const float*)d_in[2];
  const float* C_i = (const float*)d_in[3];
  const float* no  = (const float*)d_in[4];
  const float* pr  = (const float*)d_in[5];
  const float* pi  = (const float*)d_in[6];
  float* out = (float*)d_out;
  int B = in_sizes[2] / (NN * NN);
  mmse_kernel<<<B, 256, 0, stream>>>(y_r, y_i, C_r, C_i, no, pr, pi, out);
}